// GraphUNet_21792664060435
// MI455X (gfx1250) — compile-verified
//
#include <hip/hip_runtime.h>
#include <math.h>

typedef __attribute__((ext_vector_type(2))) float v2f;
typedef __attribute__((ext_vector_type(8))) float v8f;

#define THREADS 256

// ---------------------------------------------------------------- utilities
static __global__ void k_zero_f32(float* p, size_t n) {
  size_t i = (size_t)blockIdx.x * blockDim.x + threadIdx.x;
  if (i < n) p[i] = 0.0f;
}
static __global__ void k_fill_i32(int* p, int n, int v) {
  int i = blockIdx.x * blockDim.x + threadIdx.x;
  if (i < n) p[i] = v;
}
static __global__ void k_copy_i32(int* d, const int* s, int n) {
  int i = blockIdx.x * blockDim.x + threadIdx.x;
  if (i < n) d[i] = s[i];
}

// ------------------------------------------------- score = ||x_row||_2, idx
static __global__ void k_score_index(const float* __restrict__ x, int n,
                                     float* key, int* idx) {
  int wid  = (blockIdx.x * blockDim.x + threadIdx.x) >> 5;  // one wave32 / row
  int lane = threadIdx.x & 31;
  if (wid >= n) return;
  const float4* r = (const float4*)(x + (size_t)wid * 128);
  float4 v = r[lane];
  float s = v.x * v.x + v.y * v.y + v.z * v.z + v.w * v.w;
  for (int off = 16; off > 0; off >>= 1) s += __shfl_down(s, off, 32);
  if (lane == 0) { key[wid] = sqrtf(s + 1e-12f); idx[wid] = wid; }
}

// ------------------------------------- bitonic sort step (descending score,
//   ties -> smaller index first; implemented as ascending under custom "less")
static __global__ void k_bitonic(float* key, int* idx, int n, int size, int stride) {
  int i = blockIdx.x * blockDim.x + threadIdx.x;
  if (i >= n) return;
  int j = i ^ stride;
  if (j <= i) return;
  bool up = ((i & size) == 0);
  float ki = key[i], kj = key[j];
  int   ii = idx[i], ij = idx[j];
  bool less_ji = (kj > ki) || (kj == ki && ij < ii);  // elem j "smaller" = higher score
  if (less_ji == up) {
    key[i] = kj; key[j] = ki;
    idx[i] = ij; idx[j] = ii;
  }
}

static __global__ void k_scatter_map(int* mapping, const int* sidx, int k) {
  int j = blockIdx.x * blockDim.x + threadIdx.x;
  if (j < k) mapping[sidx[j]] = j;
}

// xp[j] = xprev[sidx[j]] * tanh(score[j])
static __global__ void k_pool_feat(const float* __restrict__ xprev,
                                   const int* __restrict__ sidx,
                                   const float* __restrict__ skey,
                                   float* __restrict__ xp, int k) {
  size_t i = (size_t)blockIdx.x * blockDim.x + threadIdx.x;
  if (i >= (size_t)k * 128) return;
  int j = (int)(i >> 7), c = (int)(i & 127);
  xp[i] = xprev[(size_t)sidx[j] * 128 + c] * tanhf(skey[j]);
}

static __global__ void k_relabel(int* ns, int* nd, int* nm,
                                 const int* __restrict__ ps, const int* __restrict__ pd,
                                 const int* __restrict__ pm,
                                 const int* __restrict__ mapping, int E) {
  int e = blockIdx.x * blockDim.x + threadIdx.x;
  if (e >= E) return;
  int ok = pm ? pm[e] : 1;
  int a = mapping[ps[e]];
  int b = mapping[pd[e]];
  ok = ok && (a >= 0) && (b >= 0);
  ns[e] = ok ? a : 0;
  nd[e] = ok ? b : 0;
  nm[e] = ok;
}

// ------------------------------------------------- edge mean-aggregate (sum)
static __global__ void k_aggregate(const float* __restrict__ x,
                                   const int* __restrict__ src,
                                   const int* __restrict__ dst,
                                   const int* __restrict__ mask,
                                   float* agg, float* deg, int E) {
  int wid  = (blockIdx.x * blockDim.x + threadIdx.x) >> 5;  // one wave32 / edge
  int lane = threadIdx.x & 31;
  if (wid >= E) return;
  if (mask && !mask[wid]) return;
  int s = src[wid], d = dst[wid];
  float4 v = ((const float4*)(x + (size_t)s * 128))[lane];
  float* o = agg + (size_t)d * 128 + lane * 4;
  atomicAdd(o + 0, v.x); atomicAdd(o + 1, v.y);
  atomicAdd(o + 2, v.z); atomicAdd(o + 3, v.w);
  if (lane == 0) atomicAdd(deg + d, 1.0f);
}

static __global__ void k_invdeg(const float* deg, float* inv, int n) {
  int i = blockIdx.x * blockDim.x + threadIdx.x;
  if (i < n) inv[i] = 1.0f / fmaxf(deg[i], 1.0f);
}

// -------- SAGE GEMM: out = act( (A1*rowscale) @ W1 + A2 @ W2 + bias )
// fp32 WMMA 16x16x4; one wave per 16x16 output tile; K = 128.
static __global__ void k_sage_gemm(const float* __restrict__ A1,
                                   const float* __restrict__ rowscale,
                                   const float* __restrict__ A2,
                                   const float* __restrict__ W1,
                                   const float* __restrict__ W2,
                                   const float* __restrict__ bias,
                                   float* __restrict__ out,
                                   int n, int relu) {
  int wave = (blockIdx.x * blockDim.x + threadIdx.x) >> 5;
  int lane = threadIdx.x & 31;
  int mt = wave >> 3;        // tiles along N(rows): n/16 ; 8 col-tiles (128/16)
  int nt = wave & 7;
  if (mt * 16 >= n) return;  // wave-uniform
  int l = lane & 15, half = lane >> 4;
  int arow = mt * 16 + l;          // A-fragment row this lane feeds (M = l)
  int col  = nt * 16 + l;          // B/C/D column this lane holds (N = l)
  float rs = rowscale[arow];
  const float* a1p = A1 + (size_t)arow * 128;
  const float* a2p = A2 + (size_t)arow * 128;
  v8f c = {0.f, 0.f, 0.f, 0.f, 0.f, 0.f, 0.f, 0.f};
  for (int k = 0; k < 128; k += 4) {
    int kk = k + 2 * half;         // VGPR0 holds K=kk, VGPR1 holds K=kk+1
    __builtin_prefetch(&W1[(size_t)(kk + 8) * 128 + col], 0, 1);
    __builtin_prefetch(&W2[(size_t)(kk + 8) * 128 + col], 0, 1);
    v2f a1 = { a1p[kk] * rs, a1p[kk + 1] * rs };
    v2f b1 = { W1[(size_t)kk * 128 + col], W1[(size_t)(kk + 1) * 128 + col] };
    c = __builtin_amdgcn_wmma_f32_16x16x4_f32(false, a1, false, b1,
                                              (short)0, c, false, false);
    v2f a2 = { a2p[kk], a2p[kk + 1] };
    v2f b2 = { W2[(size_t)kk * 128 + col], W2[(size_t)(kk + 1) * 128 + col] };
    c = __builtin_amdgcn_wmma_f32_16x16x4_f32(false, a2, false, b2,
                                              (short)0, c, false, false);
  }
  float bv = bias[col];
  for (int r = 0; r < 8; ++r) {
    int orow = mt * 16 + r + half * 8;   // C/D: VGPR r -> M = r (+8 for hi half)
    float v = c[r] + bv;
    if (relu) v = fmaxf(v, 0.0f);
    out[(size_t)orow * 128 + col] = v;
  }
}

// up[perm[j]] = h[j]
static __global__ void k_unpool(const float* __restrict__ h,
                                const int* __restrict__ perm,
                                float* __restrict__ up, int k) {
  size_t i = (size_t)blockIdx.x * blockDim.x + threadIdx.x;
  if (i >= (size_t)k * 128) return;
  int j = (int)(i >> 7), c = (int)(i & 127);
  up[(size_t)perm[j] * 128 + c] = h[i];
}

// ------------------------------------------------------------- loss reduces
static __global__ void k_loss_pair(const float* __restrict__ h,
                                   const float* __restrict__ ref,
                                   const int* __restrict__ perm,
                                   int k, int use_abs, float scale, float* accum) {
  __shared__ float sd[THREADS];
  int tid = threadIdx.x;
  size_t i = (size_t)blockIdx.x * blockDim.x + tid;
  float v = 0.0f;
  if (i < (size_t)k * 128) {
    int j = (int)(i >> 7), c = (int)(i & 127);
    float d = h[i] - ref[(size_t)perm[j] * 128 + c];
    v = use_abs ? fabsf(d) : d * d;
  }
  sd[tid] = v; __syncthreads();
  for (int s = THREADS / 2; s > 0; s >>= 1) {
    if (tid < s) sd[tid] += sd[tid + s];
    __syncthreads();
  }
  if (tid == 0) atomicAdd(accum, sd[0] * scale);
}

// L_FP: mean over top-k2 of (h[p2]*(1 - tanh(score)))^2
static __global__ void k_loss_fp(const float* __restrict__ h,
                                 const int* __restrict__ sidx,
                                 const float* __restrict__ skey,
                                 int k2, float scale, float* accum) {
  __shared__ float sd[THREADS];
  int tid = threadIdx.x;
  size_t i = (size_t)blockIdx.x * blockDim.x + tid;
  float v = 0.0f;
  if (i < (size_t)k2 * 128) {
    int j = (int)(i >> 7), c = (int)(i & 127);
    float d = h[(size_t)sidx[j] * 128 + c] * (1.0f - tanhf(skey[j]));
    v = d * d;
  }
  sd[tid] = v; __syncthreads();
  for (int s = THREADS / 2; s > 0; s >>= 1) {
    if (tid < s) sd[tid] += sd[tid + s];
    __syncthreads();
  }
  if (tid == 0) atomicAdd(accum, sd[0] * scale);
}

static __global__ void k_write_loss(float* out, const float* accum, size_t pos) {
  if (blockIdx.x == 0 && threadIdx.x == 0) out[pos] = accum[0];
}

// ---------------------------------------------------------------- launcher
extern "C" void kernel_launch(void* const* d_in, const int* in_sizes, int n_in,
                              void* d_out, int out_size, void* d_ws, size_t ws_size,
                              hipStream_t stream) {
  const int N = 65536, C = 128, E = 1048576;
  const int n1 = 32768, n2 = 16384, n3 = 8192;
  const size_t NC = (size_t)N * C;

  const float* x0   = (const float*)d_in[0];
  const int*   ei   = (const int*)d_in[1];
  const int*   src0 = ei;
  const int*   dst0 = ei + E;
  const float* Wl   = (const float*)d_in[2];   // (6,128,128) row-major
  const float* Wr   = (const float*)d_in[3];
  const float* bb   = (const float*)d_in[4];   // (6,128)
  float* out = (float*)d_out;

  // ---- workspace carve ----
  float* w = (float*)d_ws;
  float* agg  = w;  w += NC;
  float* tmpA = w;  w += NC;                    // xp / up buffer
  float* h1   = w;  w += (size_t)n1 * C;
  float* h2   = w;  w += (size_t)n2 * C;
  float* h3   = w;  w += (size_t)n3 * C;
  float* deg  = w;  w += N;
  float* invd = w;  w += N;
  float* skey = w;  w += N;
  float* accum = w; w += 4;
  int* ip = (int*)w;
  int* sidx    = ip; ip += N;
  int* mapping = ip; ip += N;
  int* perm0   = ip; ip += n1;
  int* perm1   = ip; ip += n2;
  int* perm2   = ip; ip += n3;
  int* s1 = ip; ip += E;  int* d1 = ip; ip += E;  int* m1 = ip; ip += E;
  int* s2 = ip; ip += E;  int* d2 = ip; ip += E;  int* m2 = ip; ip += E;
  int* s3 = ip; ip += E;  int* d3 = ip; ip += E;  int* m3 = ip; ip += E;

  auto blks = [](size_t n) { return (unsigned)((n + THREADS - 1) / THREADS); };

  auto sort_pairs = [&](float* key, int* idx, int n) {
    unsigned b = blks((size_t)n);
    for (int size = 2; size <= n; size <<= 1)
      for (int stride = size >> 1; stride > 0; stride >>= 1)
        k_bitonic<<<b, THREADS, 0, stream>>>(key, idx, n, size, stride);
  };

  // mean-SAGE conv: out_h = act( mean(agg of xin over edges) @ W1 + xin @ W2 + b )
  auto conv = [&](float* out_h, const float* xin, const int* es, const int* ed,
                  const int* em, int n, int ci, int relu) {
    k_zero_f32<<<blks((size_t)n * C), THREADS, 0, stream>>>(agg, (size_t)n * C);
    k_zero_f32<<<blks((size_t)n), THREADS, 0, stream>>>(deg, (size_t)n);
    k_aggregate<<<E / 8, THREADS, 0, stream>>>(xin, es, ed, em, agg, deg, E);
    k_invdeg<<<blks((size_t)n), THREADS, 0, stream>>>(deg, invd, n);
    k_sage_gemm<<<n / 16, THREADS, 0, stream>>>(
        agg, invd, xin,
        Wl + (size_t)ci * C * C, Wr + (size_t)ci * C * C, bb + (size_t)ci * C,
        out_h, n, relu);
  };

  k_zero_f32<<<1, THREADS, 0, stream>>>(accum, 1);

  // =============== DOWN level 0: pool x0 (65536 -> 32768), conv0 ===============
  k_score_index<<<N / 8, THREADS, 0, stream>>>(x0, N, skey, sidx);
  sort_pairs(skey, sidx, N);
  k_copy_i32<<<blks(n1), THREADS, 0, stream>>>(perm0, sidx, n1);
  k_fill_i32<<<blks(N), THREADS, 0, stream>>>(mapping, N, -1);
  k_scatter_map<<<blks(n1), THREADS, 0, stream>>>(mapping, sidx, n1);
  k_pool_feat<<<blks((size_t)n1 * C), THREADS, 0, stream>>>(x0, sidx, skey, tmpA, n1);
  k_relabel<<<blks(E), THREADS, 0, stream>>>(s1, d1, m1, src0, dst0, nullptr, mapping, E);
  conv(h1, tmpA, s1, d1, m1, n1, 0, 1);
  // losses (i=0): L_sim & L_reg both vs x0[perm0]  (xs[-1] == xs[0] at i=0)
  k_loss_pair<<<blks((size_t)n1 * C), THREADS, 0, stream>>>(h1, x0, perm0, n1, 0, 1.0f / ((float)n1 * C), accum);
  k_loss_pair<<<blks((size_t)n1 * C), THREADS, 0, stream>>>(h1, x0, perm0, n1, 1, 1.0f / ((float)n1 * C), accum);

  // =============== sort h1 scores (serves L_FP(0) AND level-1 pool) ===========
  k_score_index<<<n1 / 8, THREADS, 0, stream>>>(h1, n1, skey, sidx);
  sort_pairs(skey, sidx, n1);
  k_loss_fp<<<blks((size_t)n2 * C), THREADS, 0, stream>>>(h1, sidx, skey, n2, 1.0f / ((float)n2 * C), accum);

  // =============== DOWN level 1: pool h1 (32768 -> 16384), conv1 ==============
  k_copy_i32<<<blks(n2), THREADS, 0, stream>>>(perm1, sidx, n2);
  k_fill_i32<<<blks(n1), THREADS, 0, stream>>>(mapping, n1, -1);
  k_scatter_map<<<blks(n2), THREADS, 0, stream>>>(mapping, sidx, n2);
  k_pool_feat<<<blks((size_t)n2 * C), THREADS, 0, stream>>>(h1, sidx, skey, tmpA, n2);
  k_relabel<<<blks(E), THREADS, 0, stream>>>(s2, d2, m2, s1, d1, m1, mapping, E);
  conv(h2, tmpA, s2, d2, m2, n2, 1, 1);
  k_loss_pair<<<blks((size_t)n2 * C), THREADS, 0, stream>>>(h2, x0, perm1, n2, 0, 1.0f / ((float)n2 * C), accum);
  k_loss_pair<<<blks((size_t)n2 * C), THREADS, 0, stream>>>(h2, h1, perm1, n2, 1, 1.0f / ((float)n2 * C), accum);

  // =============== sort h2 scores (serves L_FP(1) AND level-2 pool) ===========
  k_score_index<<<n2 / 8, THREADS, 0, stream>>>(h2, n2, skey, sidx);
  sort_pairs(skey, sidx, n2);
  k_loss_fp<<<blks((size_t)n3 * C), THREADS, 0, stream>>>(h2, sidx, skey, n3, 1.0f / ((float)n3 * C), accum);

  // =============== DOWN level 2: pool h2 (16384 -> 8192), conv2 ===============
  k_copy_i32<<<blks(n3), THREADS, 0, stream>>>(perm2, sidx, n3);
  k_fill_i32<<<blks(n2), THREADS, 0, stream>>>(mapping, n2, -1);
  k_scatter_map<<<blks(n3), THREADS, 0, stream>>>(mapping, sidx, n3);
  k_pool_feat<<<blks((size_t)n3 * C), THREADS, 0, stream>>>(h2, sidx, skey, tmpA, n3);
  k_relabel<<<blks(E), THREADS, 0, stream>>>(s3, d3, m3, s2, d2, m2, mapping, E);
  conv(h3, tmpA, s3, d3, m3, n3, 2, 1);
  k_loss_pair<<<blks((size_t)n3 * C), THREADS, 0, stream>>>(h3, x0, perm2, n3, 0, 1.0f / ((float)n3 * C), accum);
  k_loss_pair<<<blks((size_t)n3 * C), THREADS, 0, stream>>>(h3, h2, perm2, n3, 1, 1.0f / ((float)n3 * C), accum);

  // =============== UP path ====================================================
  // i=2: unpool h3 -> 16384, conv idx 4 (overwrites h2, no longer needed)
  k_zero_f32<<<blks((size_t)n2 * C), THREADS, 0, stream>>>(tmpA, (size_t)n2 * C);
  k_unpool<<<blks((size_t)n3 * C), THREADS, 0, stream>>>(h3, perm2, tmpA, n3);
  conv(h2, tmpA, s2, d2, m2, n2, 4, 1);
  // i=1: unpool -> 32768, conv idx 3 (overwrites h1)
  k_zero_f32<<<blks((size_t)n1 * C), THREADS, 0, stream>>>(tmpA, (size_t)n1 * C);
  k_unpool<<<blks((size_t)n2 * C), THREADS, 0, stream>>>(h2, perm1, tmpA, n2);
  conv(h1, tmpA, s1, d1, m1, n1, 3, 1);
  // final: unpool -> 65536, conv idx 5, NO relu, write straight to d_out
  k_zero_f32<<<blks(NC), THREADS, 0, stream>>>(tmpA, NC);
  k_unpool<<<blks((size_t)n1 * C), THREADS, 0, stream>>>(h1, perm0, tmpA, n1);
  conv(out, tmpA, src0, dst0, nullptr, N, 5, 0);

  k_write_loss<<<1, 1, 0, stream>>>(out, accum, NC);
}